// RNNEI_25409026523650
// MI455X (gfx1250) — compile-verified
//
#include <hip/hip_runtime.h>
#include <hip/hip_bf16.h>
#include <math.h>

typedef __attribute__((ext_vector_type(2))) float v2f;
typedef __attribute__((ext_vector_type(8))) float v8f;

#define N_HID   300
#define N_EXC   240
#define LSEQ    200
#define BATCH   1024
#define ALPHA   0.25f
#define NSCALE  0.025f          // sigma_neu * sqrt(alpha) = 0.05 * 0.5
#define LSTR    306             // LDS row stride (floats): even (8B align) + conflict-free
#define NTILES  19              // ceil(300/16)

__global__ __launch_bounds__(256)
void rnnei_step_kernel(const float* __restrict__ xin,    // [B,L,2]
                       const float* __restrict__ h0,     // [B,300]
                       const float* __restrict__ w_in,   // [300,2]
                       const float* __restrict__ b_in,   // [300]
                       const float* __restrict__ w_hh,   // [300,300]
                       const float* __restrict__ w_out,  // [2,300]
                       const float* __restrict__ b_out,  // [2]
                       const float* __restrict__ noise,  // [L,B,300]
                       float* __restrict__ hid_out,      // [B,L,300]
                       float* __restrict__ out_out,      // [B,L,2]
                       float* __restrict__ hfin)         // [B,300]
{
    __shared__ float hLDS[16 * LSTR];   // current hidden state, 16 batch rows
    __shared__ float aLDS[16 * LSTR];   // tanh(h) * sign(col)
    __shared__ float xLDS[16 * 2];      // input signal for current t
    __shared__ float oLDS[16 * 2];      // output-head partial sums

    const int tid  = threadIdx.x;
    const int bb   = blockIdx.x * 16;   // batch base for this block
    const int wave = tid >> 5;
    const int lane = tid & 31;
    const int ln15 = lane & 15;
    const int hi   = lane >> 4;         // lane half selects K pair / M half

    // ---- init: load h0 tile into LDS, zero output accumulators ----
    for (int i = tid; i < 16 * N_HID; i += 256) {
        int m = i / N_HID, k = i - m * N_HID;
        hLDS[m * LSTR + k] = h0[(bb + m) * N_HID + k];
    }
    if (tid < 32) oLDS[tid] = 0.0f;
    __syncthreads();

    for (int t = 0; t < LSEQ; ++t) {
        // ---- phase A: act = tanh(h) * EI-sign; drain prev output; stage x_t ----
        for (int i = tid; i < 16 * N_HID; i += 256) {
            int m = i / N_HID, k = i - m * N_HID;
            float a = tanhf(hLDS[m * LSTR + k]);
            aLDS[m * LSTR + k] = (k < N_EXC) ? a : -a;
        }
        if (tid < 32) {
            int m = tid >> 1, o = tid & 1;
            if (t > 0) {
                float v = oLDS[tid] + b_out[o];
                v = fminf(fmaxf(v, -20.0f), 20.0f);
                out_out[(((long)(bb + m)) * LSEQ + (t - 1)) * 2 + o] = v;
            }
            oLDS[tid] = 0.0f;
            xLDS[tid] = xin[(((long)(bb + m)) * LSEQ + t) * 2 + o];
        }
        __syncthreads();

        // ---- phase B: per-wave 16x16 output tiles of act @ W_recT via f32 WMMA ----
        for (int tile = wave; tile < NTILES; tile += 8) {
            const int n0 = tile * 16;
            const int n  = n0 + ln15;
            const int nc = (n < N_HID) ? n : (N_HID - 1);   // clamp for loads only
            const float* wrow = w_hh + (long)nc * N_HID;    // row-major [n][k]

            v8f c = {0.f, 0.f, 0.f, 0.f, 0.f, 0.f, 0.f, 0.f};
            // K = 300 = 75 * 4 exactly
            for (int k0 = 0; k0 < N_HID; k0 += 4) {
                int kp = k0 + 2 * hi;                       // lane-half K pair
                v2f a = *(const v2f*)(aLDS + ln15 * LSTR + kp);  // A[m=ln15][kp..kp+1]
                v2f b = *(const v2f*)(wrow + kp);                // B[kp..kp+1][n]
                c = __builtin_amdgcn_wmma_f32_16x16x4_f32(
                        /*neg_a=*/false, a, /*neg_b=*/false, b,
                        /*c_mod=*/(short)0, c, /*reuse_a=*/false, /*reuse_b=*/false);
            }

            // ---- epilogue: h' = (1-a)h + a(proj + c) + ns*noise ----
            const bool  valid = (n < N_HID);
            const float wi0 = w_in[nc * 2 + 0];
            const float wi1 = w_in[nc * 2 + 1];
            const float bi  = b_in[nc];
            const float wo0 = w_out[0 * N_HID + nc];
            const float wo1 = w_out[1 * N_HID + nc];
            #pragma unroll
            for (int r = 0; r < 8; ++r) {
                int m     = r + 8 * hi;                     // C/D layout row
                int b_idx = bb + m;
                float h_old = hLDS[m * LSTR + nc];
                float proj  = wi0 * xLDS[m * 2 + 0] + wi1 * xLDS[m * 2 + 1] + bi;
                float nz    = noise[((long)t * BATCH + b_idx) * N_HID + nc];
                float hn    = (1.0f - ALPHA) * h_old + ALPHA * (proj + c[r]) + NSCALE * nz;
                if (valid) {
                    hLDS[m * LSTR + n] = hn;
                    hid_out[(((long)b_idx) * LSEQ + t) * N_HID + n] = hn;
                    atomicAdd(&oLDS[m * 2 + 0], hn * wo0);  // ds_add_f32
                    atomicAdd(&oLDS[m * 2 + 1], hn * wo1);
                }
            }
        }
        __syncthreads();
    }

    // ---- tail: last output row + final hidden state ----
    if (tid < 32) {
        int m = tid >> 1, o = tid & 1;
        float v = oLDS[tid] + b_out[o];
        v = fminf(fmaxf(v, -20.0f), 20.0f);
        out_out[(((long)(bb + m)) * LSEQ + (LSEQ - 1)) * 2 + o] = v;
    }
    for (int i = tid; i < 16 * N_HID; i += 256) {
        int m = i / N_HID, k = i - m * N_HID;
        hfin[(bb + m) * N_HID + k] = hLDS[m * LSTR + k];
    }
}

extern "C" void kernel_launch(void* const* d_in, const int* in_sizes, int n_in,
                              void* d_out, int out_size, void* d_ws, size_t ws_size,
                              hipStream_t stream) {
    (void)in_sizes; (void)n_in; (void)d_ws; (void)ws_size; (void)out_size;

    const float* xin   = (const float*)d_in[0];  // input_signal [B,L,2]
    const float* h0    = (const float*)d_in[1];  // hidden [B,300]
    /* d_in[2] = length (const int*), fixed at 200 by the reference */
    const float* w_in  = (const float*)d_in[3];  // [300,2]
    const float* b_in  = (const float*)d_in[4];  // [300]
    const float* w_hh  = (const float*)d_in[5];  // [300,300]
    const float* w_out = (const float*)d_in[6];  // [2,300]
    const float* b_out = (const float*)d_in[7];  // [2]
    const float* noise = (const float*)d_in[8];  // [L,B,300]

    float* out     = (float*)d_out;
    float* hid_out = out;                                          // [B,L,300]
    float* out_out = out + (size_t)BATCH * LSEQ * N_HID;           // [B,L,2]
    float* hfin    = out_out + (size_t)BATCH * LSEQ * 2;           // [B,300]

    dim3 grid(BATCH / 16);   // 64 independent batch-tile blocks
    dim3 block(256);         // 8 wave32s: N=300 split as 19 tiles round-robin
    rnnei_step_kernel<<<grid, block, 0, stream>>>(
        xin, h0, w_in, b_in, w_hh, w_out, b_out, noise,
        hid_out, out_out, hfin);
}